// MyConv2d_68444598829654
// MI455X (gfx1250) — compile-verified
//
#include <hip/hip_runtime.h>

typedef float v2f __attribute__((ext_vector_type(2)));
typedef float v8f __attribute__((ext_vector_type(8)));

#define CIN   64
#define COUT  128
#define HIN   112
#define WIN   112
#define HOUT  110
#define WOUT  110
#define CHW   (HIN * WIN)          // 12544
#define KTOT  576                  // CIN * 3 * 3
#define KC    16                   // K-chunk per LDS stage
#define NCHUNK (KTOT / KC)         // 36
#define BN    128                  // output pixels per block (N tile)
// Pair-interleaved LDS: element (k, col) at [(k>>1)*PSTR + 2*col + (k&1)].
// PSTR = 288: 288 mod 64 == 32 -> the two lane halves read disjoint bank sets.
#define PSTR  288
// K-order permutation: k' = tap*64 + c  (tap = kh*3+kw). A chunk of 16 k'
// never crosses a tap boundary, so per chunk tap/c0 are uniform scalars.

#if __has_builtin(__builtin_amdgcn_global_load_async_to_lds_b32)
#define USE_ASYNC 1
#else
#define USE_ASYNC 0
#endif

typedef __attribute__((address_space(1))) int g_i32;
typedef __attribute__((address_space(3))) int l_i32;

__device__ __forceinline__ g_i32* g1_b32(const void* p) {
    // flat global addresses are bit-identical to AS(1) addresses
    return (g_i32*)(uintptr_t)p;
}
__device__ __forceinline__ l_i32* l3_b32(void* p) {
    // LDS byte offset == low 32 bits of the flat shared-aperture address
    return (l_i32*)(uintptr_t)(uint32_t)(uintptr_t)p;
}

__device__ __forceinline__ void wait_async0() {
#if __has_builtin(__builtin_amdgcn_s_wait_asynccnt)
    __builtin_amdgcn_s_wait_asynccnt(0);
#else
    asm volatile("s_wait_asynccnt 0x0" ::: "memory");
#endif
}

__global__ __launch_bounds__(256)
void conv2d_implicit_gemm_wmma_f32(const float* __restrict__ X,
                                   const float* __restrict__ W,
                                   float* __restrict__ Y)
{
    __shared__ float Ap[2][8 * PSTR];   // weights chunk, pair-interleaved
    __shared__ float Bp[2][8 * PSTR];   // im2col chunk, pair-interleaved

    const int tid  = threadIdx.x;
    const int lane = tid & 31;
    const int wv   = tid >> 5;          // wave id 0..7
    const int mgrp = wv & 3;            // M group: Cout tiles 2*mgrp, 2*mgrp+1
    const int ngrp = wv >> 2;           // N group: pixel subtiles ngrp*4 .. +3
    const int half = lane >> 4;
    const int l16  = lane & 15;

    const int pixBase = blockIdx.x * BN;

    // ---- B staging mapping: element i -> (kk = 2*i + khi, col n); all static
    const int n   = tid & 127;
    const int khi = tid >> 7;
    const int p   = pixBase + n;
    const int pb  = p / (HOUT * WOUT);
    const int pr  = p - pb * (HOUT * WOUT);
    const int oh  = pr / WOUT;
    const int ow  = pr - oh * WOUT;
    // channel for element i is c = c0 + 2*i + khi  ->  fold khi into the base
    const int Xb2 = pb * (CIN * CHW) + oh * WIN + ow + khi * CHW;
    float* const bdst0 = &Bp[0][2 * n + khi];
    float* const bdst1 = &Bp[1][2 * n + khi];

    // ---- A staging mapping: thread stages k' = k0+kb .. +kb+7 for row co ----
    const int co = tid >> 1;            // 0..127 (Cout row)
    const int kb = (tid & 1) * 8;       // k' sub-offset 0 or 8
    // W element for k' is W[co*576 + c*9 + tap], c = c0 + (kb+j)
    const float* const wbase = W + co * KTOT + kb * 9;
    float* const adst0 = &Ap[0][(kb >> 1) * PSTR + 2 * co];
    float* const adst1 = &Ap[1][(kb >> 1) * PSTR + 2 * co];

    v8f acc[2][4];
    #pragma unroll
    for (int mt = 0; mt < 2; ++mt)
        #pragma unroll
        for (int nt = 0; nt < 4; ++nt) acc[mt][nt] = {};

#if !USE_ASYNC
    float wa[8], xv[8];
#endif

    // ---------------- staging helpers ----------------
    #define STAGE_LOADS(K0, ADST, BDST)                                        \
    {                                                                          \
        const int tap = (K0) >> 6;                                             \
        const int c0  = (K0) & 63;                                             \
        const int kh  = (tap >= 6) ? 2 : ((tap >= 3) ? 1 : 0);                 \
        const int kw  = tap - 3 * kh;                                          \
        const float* wsrc = wbase + c0 * 9 + tap;                              \
        const float* xsrc = X + Xb2 + c0 * CHW + kh * WIN + kw;                \
        _Pragma("unroll")                                                      \
        for (int j = 0; j < 8; ++j)                                            \
            __builtin_amdgcn_global_load_async_to_lds_b32(                     \
                g1_b32(wsrc + 9 * j),                                          \
                l3_b32((ADST) + (j >> 1) * PSTR + (j & 1)), 0, 0);             \
        _Pragma("unroll")                                                      \
        for (int i = 0; i < 8; ++i)                                            \
            __builtin_amdgcn_global_load_async_to_lds_b32(                     \
                g1_b32(xsrc + 2 * CHW * i),                                    \
                l3_b32((BDST) + i * PSTR), 0, 0);                              \
    }

    #define STAGE_REGS(K0)                                                     \
    {                                                                          \
        const int tap = (K0) >> 6;                                             \
        const int c0  = (K0) & 63;                                             \
        const int kh  = (tap >= 6) ? 2 : ((tap >= 3) ? 1 : 0);                 \
        const int kw  = tap - 3 * kh;                                          \
        const float* wsrc = wbase + c0 * 9 + tap;                              \
        const float* xsrc = X + Xb2 + c0 * CHW + kh * WIN + kw;                \
        _Pragma("unroll")                                                      \
        for (int j = 0; j < 8; ++j) wa[j] = wsrc[9 * j];                       \
        _Pragma("unroll")                                                      \
        for (int i = 0; i < 8; ++i) xv[i] = xsrc[2 * CHW * i];                 \
    }

    #define STORE_REGS(ADST, BDST)                                             \
    {                                                                          \
        _Pragma("unroll")                                                      \
        for (int j = 0; j < 8; ++j) (ADST)[(j >> 1) * PSTR + (j & 1)] = wa[j]; \
        _Pragma("unroll")                                                      \
        for (int i = 0; i < 8; ++i) (BDST)[i * PSTR] = xv[i];                  \
    }

    // ---------------- prologue: stage chunk 0 into buffer 0 ----------------
#if USE_ASYNC
    STAGE_LOADS(0, adst0, bdst0)
    wait_async0();
#else
    STAGE_REGS(0)
    STORE_REGS(adst0, bdst0)
#endif
    __syncthreads();

    // ---------------- main loop: double-buffered LDS ----------------
    int cur = 0;
    for (int kc = 0; kc < NCHUNK; ++kc) {
        const bool more = (kc + 1 < NCHUNK);

        if (more) {
            const int k0 = (kc + 1) * KC;
#if USE_ASYNC
            if (cur == 0) STAGE_LOADS(k0, adst1, bdst1)
            else          STAGE_LOADS(k0, adst0, bdst0)
#else
            STAGE_REGS(k0)
#endif
        }

        // ---- compute: 2M x 4N wave tile; 6 LDS loads per 8 WMMAs ----
        {
            const float* apan = &Ap[cur][half * PSTR + 2 * (mgrp * 32 + l16)];
            const float* bpan = &Bp[cur][half * PSTR + 2 * (ngrp * 64 + l16)];
            v2f afrag[2][2];
            v2f bfrag[2][4];
            #pragma unroll
            for (int mt = 0; mt < 2; ++mt)
                afrag[0][mt] = *(const v2f*)&apan[32 * mt];
            #pragma unroll
            for (int nt = 0; nt < 4; ++nt)
                bfrag[0][nt] = *(const v2f*)&bpan[32 * nt];

            #pragma unroll
            for (int s = 0; s < 4; ++s) {
                const int cp = s & 1;
                const int np = cp ^ 1;
                if (s < 3) {   // prefetch next k-step's fragments
                    const int row = 2 * (s + 1) * PSTR;
                    #pragma unroll
                    for (int mt = 0; mt < 2; ++mt)
                        afrag[np][mt] = *(const v2f*)&apan[row + 32 * mt];
                    #pragma unroll
                    for (int nt = 0; nt < 4; ++nt)
                        bfrag[np][nt] = *(const v2f*)&bpan[row + 32 * nt];
                }
                #pragma unroll
                for (int mt = 0; mt < 2; ++mt)
                    #pragma unroll
                    for (int nt = 0; nt < 4; ++nt)
                        acc[mt][nt] = __builtin_amdgcn_wmma_f32_16x16x4_f32(
                            false, afrag[cp][mt], false, bfrag[cp][nt],
                            (short)0, acc[mt][nt], false, false);
            }
        }

        if (more) {
#if USE_ASYNC
            wait_async0();
#else
            if (cur == 0) STORE_REGS(adst1, bdst1)
            else          STORE_REGS(adst0, bdst0)
#endif
        }
        __syncthreads();
        cur ^= 1;
    }

    // ---- store: C/D 16x16 layout: VGPR r -> M = r + 8*half, N = l16 ----
    #pragma unroll
    for (int mt = 0; mt < 2; ++mt) {
        const int coutBase = (2 * mgrp + mt) * 16 + 8 * half;
        #pragma unroll
        for (int nt = 0; nt < 4; ++nt) {
            const int sp  = pixBase + ngrp * 64 + nt * 16 + l16;
            const int sb  = sp / (HOUT * WOUT);
            const int spr = sp - sb * (HOUT * WOUT);
            const int soh = spr / WOUT;
            const int sow = spr - soh * WOUT;
            #pragma unroll
            for (int r = 0; r < 8; ++r) {
                Y[((sb * COUT + (coutBase + r)) * HOUT + soh) * WOUT + sow] =
                    acc[mt][nt][r];
            }
        }
    }
}

extern "C" void kernel_launch(void* const* d_in, const int* in_sizes, int n_in,
                              void* d_out, int out_size, void* d_ws, size_t ws_size,
                              hipStream_t stream) {
    const float* X = (const float*)d_in[0];   // (32,64,112,112) f32
    const float* W = (const float*)d_in[1];   // (128,64,3,3)    f32
    float* Y = (float*)d_out;                 // (32,128,110,110) f32

    const int totalPix = 32 * HOUT * WOUT;    // 387200 = 128 * 3025
    const int grid = totalPix / BN;           // 3025 blocks
    conv2d_implicit_gemm_wmma_f32<<<grid, 256, 0, stream>>>(X, W, Y);
}